// MeanSquaredError2__87222195847404
// MI455X (gfx1250) — compile-verified
//
#include <hip/hip_runtime.h>
#include <math.h>

#define NJJ 14
#define COLN 14
#define HW 196            // 14*14
#define WAVES_PER_BLOCK 8

typedef float v2f __attribute__((ext_vector_type(2)));
typedef float v8f __attribute__((ext_vector_type(8)));

// Workspace layout
struct Ws {
    double acc[4];            // d1sum, d2sum, nsum, pad
    float  G[HW];             // G[r*14 + y] : filter response at row r for delta at y
    float  cmin[COLN], cmax[COLN], csum[COLN], csumsq[COLN];
};

// ---------------------------------------------------------------------------
// Init: build the 1D-delta Gaussian response matrix G (symmetric padding,
// radius 4, sigma 1, f64 kernel weights cast to f32 like the reference),
// its per-column stats, and zero the global accumulators.
// ---------------------------------------------------------------------------
__global__ void init_kernel(Ws* ws) {
    if (blockIdx.x != 0 || threadIdx.x != 0) return;
    ws->acc[0] = 0.0; ws->acc[1] = 0.0; ws->acc[2] = 0.0; ws->acc[3] = 0.0;

    double kd[9]; double s = 0.0;
    for (int i = 0; i < 9; ++i) { double x = (double)(i - 4); kd[i] = exp(-0.5 * x * x); s += kd[i]; }
    float kf[9];
    for (int i = 0; i < 9; ++i) kf[i] = (float)(kd[i] / s);

    for (int r = 0; r < COLN; ++r) {
        for (int y = 0; y < COLN; ++y) {
            float a = 0.f;
            for (int i = 0; i < 9; ++i) {
                int p = r + i;  // padded index in [0,22)
                int o = (p < 4) ? (3 - p) : ((p < 18) ? (p - 4) : (31 - p)); // symmetric pad
                if (o == y) a += kf[i];
            }
            ws->G[r * COLN + y] = a;
        }
    }
    for (int y = 0; y < COLN; ++y) {
        float mn = 3.4e38f, mx = -3.4e38f, su = 0.f, sq = 0.f;
        for (int r = 0; r < COLN; ++r) {
            float g = ws->G[r * COLN + y];
            mn = fminf(mn, g); mx = fmaxf(mx, g); su += g; sq += g * g;
        }
        ws->cmin[y] = mn; ws->cmax[y] = mx; ws->csum[y] = su; ws->csumsq[y] = sq;
    }
}

// ---------------------------------------------------------------------------
// Main: one wave32 per joint. Streams h once; WMMA computes U = H*G (f32);
// VALU computes sums/argmax from the same registers; lane0 does decode+gather.
// ---------------------------------------------------------------------------
__global__ __launch_bounds__(256) void pose_loss_kernel(
    const float* __restrict__ osp,   // [B, 2*NJ, 196]
    const float* __restrict__ h,     // [B, NJ, 196]
    const float* __restrict__ t,     // [B, NJ, 2]
    const float* __restrict__ v,     // [B, NJ, 2]
    Ws* __restrict__ ws,
    int total_joints) {

    __shared__ float sG[HW];
    __shared__ float sMin[COLN], sMax[COLN], sSum[COLN], sSq[COLN];
    __shared__ float wpart[WAVES_PER_BLOCK][3];

    const int tid = threadIdx.x;
    if (tid < HW)                       sG[tid] = ws->G[tid];
    else if (tid < HW + COLN)           sMin[tid - HW] = ws->cmin[tid - HW];
    else if (tid < HW + 2 * COLN)       sMax[tid - HW - COLN] = ws->cmax[tid - HW - COLN];
    else if (tid < HW + 3 * COLN)       sSum[tid - HW - 2 * COLN] = ws->csum[tid - HW - 2 * COLN];
    else if (tid < HW + 4 * COLN)       sSq[tid - HW - 3 * COLN] = ws->csumsq[tid - HW - 3 * COLN];
    __syncthreads();

    const int wave = tid >> 5;
    const int lane = tid & 31;
    int joint = blockIdx.x * WAVES_PER_BLOCK + wave;
    const bool active = joint < total_joints;
    if (!active) joint = total_joints - 1;   // clamp; contribution zeroed later

    const int b = joint / NJJ;
    const int j = joint - b * NJJ;

    // per-joint metadata (uniform loads)
    const float t0 = t[(size_t)joint * 2 + 0];
    const float t1 = t[(size_t)joint * 2 + 1];
    const float v0 = v[(size_t)joint * 2 + 0];
    const bool  vis = v0 > 0.5f;
    int xi = (int)(t0 * 14.0f); xi = xi < 0 ? 0 : (xi > 13 ? 13 : xi);
    int yi = (int)(t1 * 14.0f); yi = yi < 0 ? 0 : (yi > 13 ? 13 : yi);

    const int row = lane & 15;     // M index (A) and N index (B/C)
    const int hl  = lane >> 4;     // lane-half

    // B-matrix regs: B[k][n] = G[k][n] in (inferred) 4x16 f32 WMMA layout.
    v2f bm[4];
#pragma unroll
    for (int s = 0; s < 4; ++s) {
        const int k0 = 4 * s + 2 * hl;
        bm[s].x = (k0     < COLN && row < COLN) ? sG[k0 * COLN + row]       : 0.f;
        bm[s].y = (k0 + 1 < COLN && row < COLN) ? sG[(k0 + 1) * COLN + row] : 0.f;
    }

    // A-matrix regs: slices of H (this joint's 14x14 heatmap), zero-padded to 16x16.
    const float* hb = h + (size_t)joint * HW;
    float shh = 0.f, sh = 0.f, r0sq = 0.f;
    float mval = -1.0f; int midx = 0;
    v2f am[4];
#pragma unroll
    for (int s = 0; s < 4; ++s) {
        const int c0 = 4 * s + 2 * hl;
        const bool aval = (row < COLN) && (c0 < COLN);      // c0 even -> c0+1 valid too
        const float* p = aval ? (hb + row * COLN + c0) : hb; // in-range address always
        v2f av = *(const v2f*)p;                             // global_load_b64
        if (!aval) { av.x = 0.f; av.y = 0.f; }
        am[s] = av;
        // scalar stats off the same registers
        shh += av.x * av.x + av.y * av.y;
        sh  += av.x + av.y;
        if (row == 0) r0sq += av.x * av.x + av.y * av.y;
        const int ix = row * COLN + c0;
        if (aval && av.x > mval) { mval = av.x; midx = ix; }
        if (aval && av.y > mval) { mval = av.y; midx = ix + 1; }
    }

    // U = H * G on the matrix pipe: 4 x v_wmma_f32_16x16x4_f32 (K = 16 padded)
    v8f u = {0.f, 0.f, 0.f, 0.f, 0.f, 0.f, 0.f, 0.f};
#pragma unroll
    for (int s = 0; s < 4; ++s) {
        u = __builtin_amdgcn_wmma_f32_16x16x4_f32(
                false, am[s], false, bm[s], (short)0, u, false, false);
    }

    // S_hg = G[:,yi]^T * U[:,xi] ; column xi of U lives in lanes {xi, xi+16}
    float shg = 0.f;
    if (row == xi) {
#pragma unroll
        for (int p = 0; p < 8; ++p) {
            const int m = p + 8 * hl;
            const float gm = (m < COLN) ? sG[m * COLN + yi] : 0.f;
            shg += u[p] * gm;
        }
    }

    // wave32 butterfly reductions (sum x4 + argmax, first-occurrence tiebreak)
#pragma unroll
    for (int off = 16; off > 0; off >>= 1) {
        shh += __shfl_xor(shh, off);
        sh  += __shfl_xor(sh,  off);
        r0sq += __shfl_xor(r0sq, off);
        shg += __shfl_xor(shg, off);
        const float om = __shfl_xor(mval, off);
        const int   oi = __shfl_xor(midx, off);
        if (om > mval || (om == mval && oi < midx)) { mval = om; midx = oi; }
    }

    if (lane == 0) {
        float d1c, nc;
        if (vis) {
            const float mx  = sMax[yi] * sMax[xi];
            const float mn  = sMin[yi] * sMin[xi];
            const float Sg1 = sSum[yi] * sSum[xi];
            const float Sg2 = sSq[yi]  * sSq[xi];
            const float A   = 1.0f / (mx - mn);
            d1c = shh - 2.0f * A * (shg - mn * sh)
                      + A * A * (Sg2 - 2.0f * mn * Sg1 + 196.0f * mn * mn);
            nc = 1.0f;
        } else {
            d1c = shh - r0sq;   // row 0 of diff zeroed for invisible joints
            nc = 0.0f;
        }
        // coordinate decode + os_ gather
        const bool cond = mval > 0.5f;
        const int  yC = midx / COLN;
        const int  xC = midx - yC * COLN;
        const float* ob = osp + ((size_t)b * (2 * NJJ) + j) * HW;
        const float osx = ob[midx];
        const float osy = ob[(size_t)NJJ * HW + midx];
        const float sc = 1.0f / 14.0f;
        const float x0 = cond ? (osx + (float)xC) * sc : 0.f;
        const float x1 = cond ? (osy + (float)yC) * sc : 0.f;
        const float e0 = (x0 - t0) * v0;
        const float e1 = (x1 - t1) * v0;
        const float d2c = e0 * e0 + e1 * e1;

        wpart[wave][0] = active ? d1c : 0.f;
        wpart[wave][1] = active ? d2c : 0.f;
        wpart[wave][2] = active ? nc  : 0.f;
    }
    __syncthreads();
    if (tid == 0) {
        double s0 = 0.0, s1 = 0.0, s2 = 0.0;
        for (int w = 0; w < WAVES_PER_BLOCK; ++w) {
            s0 += (double)wpart[w][0];
            s1 += (double)wpart[w][1];
            s2 += (double)wpart[w][2];
        }
        atomicAdd(&ws->acc[0], s0);   // global_atomic_add_f64
        atomicAdd(&ws->acc[1], s1);
        atomicAdd(&ws->acc[2], s2);
    }
}

__global__ void finalize_kernel(const Ws* __restrict__ ws, float* __restrict__ out) {
    if (blockIdx.x == 0 && threadIdx.x == 0) {
        out[0] = (float)((ws->acc[0] + ws->acc[1]) / ws->acc[2]);
    }
}

// ---------------------------------------------------------------------------
extern "C" void kernel_launch(void* const* d_in, const int* in_sizes, int n_in,
                              void* d_out, int out_size, void* d_ws, size_t ws_size,
                              hipStream_t stream) {
    const float* osp = (const float*)d_in[0];  // os_
    const float* h   = (const float*)d_in[1];  // h
    // d_in[2] = op, unused by the reference
    const float* t   = (const float*)d_in[3];  // t
    const float* v   = (const float*)d_in[4];  // v
    float* out = (float*)d_out;
    Ws* ws = (Ws*)d_ws;

    const int total_joints = in_sizes[1] / HW;                 // B * NJ
    const int blocks = (total_joints + WAVES_PER_BLOCK - 1) / WAVES_PER_BLOCK;

    init_kernel<<<1, 64, 0, stream>>>(ws);
    pose_loss_kernel<<<blocks, 256, 0, stream>>>(osp, h, t, v, ws, total_joints);
    finalize_kernel<<<1, 32, 0, stream>>>(ws, out);
}